// LocallyConnected2dV1_3006477107302
// MI455X (gfx1250) — compile-verified
//
#include <hip/hip_runtime.h>
#include <hip/hip_bf16.h>

typedef __attribute__((ext_vector_type(2))) float v2f;
typedef __attribute__((ext_vector_type(8))) float v8f;

// Problem constants (from reference)
#define NB   16      // batch
#define IC   3
#define IH   128
#define IW   128
#define KH   8
#define KW   8
#define SH   2
#define SW   2
#define NH   61
#define NW   61
#define OC   64
#define KDIM 192     // IC*KH*KW
#define LDA  196     // padded LDS row stride (floats); 196 % 64 = 4 -> conflict-free strided reads
#define OUTW 488     // 8*61

// One async b128 transfer: moves 16B/lane global->LDS, offset applied to BOTH addresses.
#define ASYNC_A(OFS)                                                           \
    asm volatile("global_load_async_to_lds_b128 %0, %1, %2 offset:" #OFS       \
                 :: "v"(ldsa), "v"(goff), "s"(wbase) : "memory")

#define ASYNC_B(OFS)                                                           \
    asm volatile("global_load_async_to_lds_b64 %0, %1, %2 offset:" #OFS        \
                 :: "v"(ldsb), "v"(xoff), "s"(xbase) : "memory")

__global__ __launch_bounds__(128) void lc2d_wmma_kernel(
    const float* __restrict__ x,      // (16,3,128,128)
    const float* __restrict__ W,      // (61,61,64,3,8,8)
    const float* __restrict__ bias,   // (64,61,61)
    float* __restrict__ out)          // (16,1,488,488)
{
    const int w = blockIdx.x;   // 0..60
    const int h = blockIdx.y;   // 0..60
    const int tid = threadIdx.x;  // 0..127 (4 waves of 32)

    __shared__ float ldsA[OC * LDA];   // W[h,w] row-major 64x192, padded -> 50176 B
    __shared__ float ldsB[NB * LDA];   // patches transposed: [batch][k]   -> 12544 B

    // ---- Stage A: DMA weight slab (contiguous 64*192 f32) straight into LDS ----
    // 2 threads per OC row; each covers a 96-float half-row with 24 x b128 asyncs.
    {
        const unsigned long long wbase =
            (unsigned long long)(const void*)W +
            (unsigned long long)(h * NW + w) * (OC * KDIM) * sizeof(float);
        const int row  = tid >> 1;          // 0..63
        const int half = tid & 1;           // 0 or 1
        const unsigned goff = (unsigned)((row * KDIM + half * 96) * sizeof(float));
        const unsigned ldsa = (unsigned)(uintptr_t)&ldsA[row * LDA + half * 96];
        ASYNC_A(0);   ASYNC_A(16);  ASYNC_A(32);  ASYNC_A(48);
        ASYNC_A(64);  ASYNC_A(80);  ASYNC_A(96);  ASYNC_A(112);
        ASYNC_A(128); ASYNC_A(144); ASYNC_A(160); ASYNC_A(176);
        ASYNC_A(192); ASYNC_A(208); ASYNC_A(224); ASYNC_A(240);
        ASYNC_A(256); ASYNC_A(272); ASYNC_A(288); ASYNC_A(304);
        ASYNC_A(320); ASYNC_A(336); ASYNC_A(352); ASYNC_A(368);
    }

    // ---- Stage B: DMA patches into LDS as B^T[b][k], k = c*64 + i*8 + j ----
    // 384 (b,c,i) tasks; each task = 8 contiguous floats (8B-aligned) = 4 x b64 asyncs.
    {
        const unsigned long long xbase = (unsigned long long)(const void*)x;
        #pragma unroll
        for (int it = 0; it < 3; ++it) {
            int task = tid + it * 128;
            int b    = task / 24;
            int rem  = task - b * 24;
            int c    = rem >> 3;
            int i    = rem & 7;
            const unsigned xoff = (unsigned)(((((b * IC + c) * IH) + (h * SH + i)) * IW
                                              + w * SW) * sizeof(float));
            const unsigned ldsb = (unsigned)(uintptr_t)&ldsB[b * LDA + (c * 64 + i * 8)];
            ASYNC_B(0); ASYNC_B(8); ASYNC_B(16); ASYNC_B(24);
        }
    }

    // Drain this wave's async fills, then rendezvous with the other waves.
    asm volatile("s_wait_asynccnt 0x0" ::: "memory");
    __syncthreads();

    // ---- Stage C: 16x16x192 per wave via 48 x V_WMMA_F32_16X16X4_F32 ----
    const int wave = tid >> 5;          // 0..3 -> OC tile
    const int lane = tid & 31;
    const int hi   = lane >> 4;         // 0: K={k0,k0+1}, 1: K={k0+2,k0+3}
    const int lo   = lane & 15;         // A: row within tile; B: batch column

    const float* ap = &ldsA[(wave * 16 + lo) * LDA + 2 * hi];
    const float* bp = &ldsB[lo * LDA + 2 * hi];

    v8f acc = {0.f, 0.f, 0.f, 0.f, 0.f, 0.f, 0.f, 0.f};
    #pragma unroll
    for (int k0 = 0; k0 < KDIM; k0 += 4) {
        v2f a = *(const v2f*)(ap + k0);
        v2f b = *(const v2f*)(bp + k0);
        acc = __builtin_amdgcn_wmma_f32_16x16x4_f32(
            /*neg_a=*/false, a, /*neg_b=*/false, b,
            /*c_mod=*/(short)0, acc, /*reuse_a=*/false, /*reuse_b=*/false);
    }

    // ---- Stage D: bias + pixel-shuffle scatter ----
    // C layout: VGPR r on lane -> (M = r + 8*hi, N = lo); oc = wave*16 + M; batch = lo
    const int batch  = lo;
    const int ocBase = wave * 16 + 8 * hi;
    float* outp = out + (size_t)batch * (OUTW * OUTW);
    #pragma unroll
    for (int r = 0; r < 8; ++r) {
        int oc = ocBase + r;
        float v = acc[r] + bias[(size_t)oc * (NH * NW) + h * NW + w];
        outp[(size_t)(h * 8 + (oc >> 3)) * OUTW + (w * 8 + (oc & 7))] = v;
    }
}

extern "C" void kernel_launch(void* const* d_in, const int* in_sizes, int n_in,
                              void* d_out, int out_size, void* d_ws, size_t ws_size,
                              hipStream_t stream) {
    const float* x    = (const float*)d_in[0];
    const float* W    = (const float*)d_in[1];
    const float* bias = (const float*)d_in[2];
    float* out = (float*)d_out;

    dim3 grid(NW, NH);   // one workgroup per output spatial position
    dim3 block(128);     // 4 waves: one 16x16 C tile each
    lc2d_wmma_kernel<<<grid, block, 0, stream>>>(x, W, bias, out);
}